// LinearBlock_75153337745552
// MI455X (gfx1250) — compile-verified
//
#include <hip/hip_runtime.h>
#include <math.h>

// ---------------------------------------------------------------------------
// Transformer block fwd + JVP for MI455X (gfx1250), fp32 via V_WMMA_F32_16X16X4_F32
// Block tile 128x128, wave tile 32x64 (2x4 WMMA subtiles), KC=16 staged in LDS.
// ---------------------------------------------------------------------------

typedef float v2f __attribute__((ext_vector_type(2)));
typedef float v8f __attribute__((ext_vector_type(8)));

#define BDIM   4
#define NSEQ   785
#define DMODEL 1024
#define HID    4096
#define NH     16
#define HD     64
#define MROWS  (BDIM * NSEQ)   // 3140

#define MT 128   // workgroup M tile
#define NT 128   // workgroup N tile
#define KC 16    // K chunk staged in LDS
#define LP 20    // LDS pitch: mult-of-4 (16B rows) and 20*k mod 64 distinct for k<16

// Generic WMMA GEMM:
//   C[z,m,n] = alpha * ( A1[z,m,:] . W1[z,n,:]  (+ A2[z,m,:] . W2[z,n,:]) )
//              + bias[n] + Res[z,m,n]
// A element addr : A + zb*a_bb + zh*a_bh + m*a_rs + k          (k-stride always 1)
// W element addr : W + zb*w_bb + zh*w_bh + n*w_ns + k*w_ks
// C element addr : C + zb*c_bb + zh*c_bh + m*c_rs + n
// z = blockIdx.z ; zb = z / Hdiv ; zh = z % Hdiv
// a_v4 / w_v4: host-asserted float4-load legality (16B alignment + K % KC == 0)
template <bool DUAL>
__global__ __launch_bounds__(256)
void gemm_f32_wmma(float* __restrict__ C,
                   const float* __restrict__ A1, const float* __restrict__ W1,
                   const float* __restrict__ A2, const float* __restrict__ W2,
                   const float* __restrict__ bias, const float* __restrict__ Res,
                   int M, int Nc, int K,
                   long a_rs, long a_bb, long a_bh,
                   long w_ns, long w_ks, long w_bb, long w_bh,
                   long c_rs, long c_bb, long c_bh,
                   long r_rs, long r_bb, long r_bh,
                   int Hdiv, float alpha, int a_v4, int w_v4)
{
    __shared__ float As1[MT * LP];
    __shared__ float As2[DUAL ? MT * LP : 4];
    __shared__ float Ws1[NT * LP];
    __shared__ float Ws2[DUAL ? NT * LP : 4];

    const int tid  = threadIdx.x;
    const int lane = tid & 31;
    const int wv   = tid >> 5;       // wave 0..7
    const int wm   = wv >> 1;        // 0..3 -> 32-row strip
    const int wn   = wv & 1;         // 0..1 -> 64-col strip
    const int half = lane >> 4;      // K-pair select / M 0..7 vs 8..15
    const int lr   = lane & 15;

    const int z  = blockIdx.z;
    const int zb = z / Hdiv, zh = z % Hdiv;
    const long aB = (long)zb * a_bb + (long)zh * a_bh;
    const long wB = (long)zb * w_bb + (long)zh * w_bh;
    const long cB = (long)zb * c_bb + (long)zh * c_bh;
    const long rB = (long)zb * r_bb + (long)zh * r_bh;

    const int mRow0 = blockIdx.y * MT;
    const int nCol0 = blockIdx.x * NT;

    v8f acc[2][4];
    const v8f vzero = {0.f, 0.f, 0.f, 0.f, 0.f, 0.f, 0.f, 0.f};
#pragma unroll
    for (int i = 0; i < 2; ++i)
#pragma unroll
        for (int j = 0; j < 4; ++j) acc[i][j] = vzero;

    for (int k0 = 0; k0 < K; k0 += KC) {
        // ---- stage A tile (k contiguous in memory) ----
        if (a_v4) {
            for (int idx = tid; idx < MT * KC / 4; idx += 256) {
                int kk4 = (idx & 3) * 4;
                int mm  = idx >> 2;
                int gm  = mRow0 + mm;
                long off = aB + (long)gm * a_rs + (k0 + kk4);
                float4 v  = {0.f, 0.f, 0.f, 0.f};
                float4 v2 = {0.f, 0.f, 0.f, 0.f};
                if (gm < M) {
                    v = *(const float4*)(A1 + off);
                    if (DUAL) v2 = *(const float4*)(A2 + off);
                    if (k0 + KC < K) __builtin_prefetch(A1 + off + KC, 0, 1);
                }
                *(float4*)&As1[mm * LP + kk4] = v;
                if (DUAL) *(float4*)&As2[mm * LP + kk4] = v2;
            }
        } else {
            for (int idx = tid; idx < MT * KC; idx += 256) {
                int kk = idx & (KC - 1);
                int mm = idx >> 4;
                int gm = mRow0 + mm, gk = k0 + kk;
                bool in = (gm < M) && (gk < K);
                long off = aB + (long)gm * a_rs + gk;
                As1[mm * LP + kk] = in ? A1[off] : 0.f;
                if (DUAL) As2[mm * LP + kk] = in ? A2[off] : 0.f;
            }
        }
        // ---- stage W tile ----
        if (w_ks == 1) {
            if (w_v4) {
                for (int idx = tid; idx < NT * KC / 4; idx += 256) {
                    int kk4 = (idx & 3) * 4;
                    int nn  = idx >> 2;
                    int gn  = nCol0 + nn;
                    long off = wB + (long)gn * w_ns + (k0 + kk4);
                    float4 v  = {0.f, 0.f, 0.f, 0.f};
                    float4 v2 = {0.f, 0.f, 0.f, 0.f};
                    if (gn < Nc) {
                        v = *(const float4*)(W1 + off);
                        if (DUAL) v2 = *(const float4*)(W2 + off);
                        if (k0 + KC < K) __builtin_prefetch(W1 + off + KC, 0, 1);
                    }
                    *(float4*)&Ws1[nn * LP + kk4] = v;
                    if (DUAL) *(float4*)&Ws2[nn * LP + kk4] = v2;
                }
            } else {
                for (int idx = tid; idx < NT * KC; idx += 256) {
                    int kk = idx & (KC - 1);
                    int nn = idx >> 4;
                    int gn = nCol0 + nn, gk = k0 + kk;
                    bool in = (gn < Nc) && (gk < K);
                    long off = wB + (long)gn * w_ns + gk;
                    Ws1[nn * LP + kk] = in ? W1[off] : 0.f;
                    if (DUAL) Ws2[nn * LP + kk] = in ? W2[off] : 0.f;
                }
            }
        } else {  // n contiguous (A@V case): n-fast mapping, coalesced over n
            for (int idx = tid; idx < NT * KC; idx += 256) {
                int nn = idx & (NT - 1);
                int kk = idx >> 7;
                int gn = nCol0 + nn, gk = k0 + kk;
                bool in = (gn < Nc) && (gk < K);
                Ws1[nn * LP + kk] = in ? W1[wB + (long)gn * w_ns + (long)gk * w_ks] : 0.f;
                if (DUAL) Ws2[nn * LP + kk] = in ? W2[wB + (long)gn * w_ns + (long)gk * w_ks] : 0.f;
            }
        }
        __syncthreads();

        const int m0 = wm * 32, n0 = wn * 64;
#pragma unroll
        for (int kk = 0; kk < KC; kk += 4) {
            // A frag 16x4 f32: lanes0-15 -> (K0,K1), lanes16-31 -> (K2,K3)
            v2f a1[2], b1[4];
#pragma unroll
            for (int i = 0; i < 2; ++i) {
                int p = (m0 + 16 * i + lr) * LP + kk + 2 * half;
                a1[i].x = As1[p];
                a1[i].y = As1[p + 1];
            }
#pragma unroll
            for (int j = 0; j < 4; ++j) {
                int p = (n0 + 16 * j + lr) * LP + kk + 2 * half;
                b1[j].x = Ws1[p];
                b1[j].y = Ws1[p + 1];
            }
#pragma unroll
            for (int i = 0; i < 2; ++i)
#pragma unroll
                for (int j = 0; j < 4; ++j)
                    acc[i][j] = __builtin_amdgcn_wmma_f32_16x16x4_f32(
                        false, a1[i], false, b1[j], (short)0, acc[i][j], false, false);

            if (DUAL) {
                v2f a2[2], b2[4];
#pragma unroll
                for (int i = 0; i < 2; ++i) {
                    int p = (m0 + 16 * i + lr) * LP + kk + 2 * half;
                    a2[i].x = As2[p];
                    a2[i].y = As2[p + 1];
                }
#pragma unroll
                for (int j = 0; j < 4; ++j) {
                    int p = (n0 + 16 * j + lr) * LP + kk + 2 * half;
                    b2[j].x = Ws2[p];
                    b2[j].y = Ws2[p + 1];
                }
#pragma unroll
                for (int i = 0; i < 2; ++i)
#pragma unroll
                    for (int j = 0; j < 4; ++j)
                        acc[i][j] = __builtin_amdgcn_wmma_f32_16x16x4_f32(
                            false, a2[i], false, b2[j], (short)0, acc[i][j], false, false);
            }
        }
        __syncthreads();
    }

    // ---- write C (f32 C layout: lanes0-15 rows 0..7, lanes16-31 rows 8..15) ----
    const int m0g = mRow0 + wm * 32, n0g = nCol0 + wn * 64;
#pragma unroll
    for (int i = 0; i < 2; ++i) {
#pragma unroll
        for (int j = 0; j < 4; ++j) {
            int col = n0g + 16 * j + lr;
            float bb = (bias != nullptr && col < Nc) ? bias[col] : 0.f;
#pragma unroll
            for (int r = 0; r < 8; ++r) {
                int row = m0g + 16 * i + 8 * half + r;
                if (row < M && col < Nc) {
                    float v = alpha * acc[i][j][r] + bb;
                    if (Res) v += Res[rB + (long)row * r_rs + col];
                    C[cB + (long)row * c_rs + col] = v;
                }
            }
        }
    }
}

// ---------------- LayerNorm forward + JVP (row per block, D=1024) ----------
__global__ __launch_bounds__(256)
void ln_jvp(const float* __restrict__ x, const float* __restrict__ xj,
            const float* __restrict__ w, const float* __restrict__ b,
            const float* __restrict__ lw, const float* __restrict__ lb,
            float* __restrict__ out, float* __restrict__ outj)
{
    const int row = blockIdx.x;
    const int tid = threadIdx.x;
    const int D = DMODEL;
    const float* xr  = x  + (long)row * D;
    const float* xjr = xj + (long)row * D;

    float lx[4], lxj[4];
    float sx = 0.f, sxj = 0.f, sxx = 0.f, sxxj = 0.f;
#pragma unroll
    for (int t = 0; t < 4; ++t) {
        int i = tid + t * 256;
        float a = xr[i], aj = xjr[i];
        lx[t] = a; lxj[t] = aj;
        sx += a; sxj += aj; sxx += a * a; sxxj += a * aj;
    }
    __shared__ float red[4][256];
    red[0][tid] = sx; red[1][tid] = sxj; red[2][tid] = sxx; red[3][tid] = sxxj;
    __syncthreads();
    for (int s = 128; s > 0; s >>= 1) {
        if (tid < s)
#pragma unroll
            for (int q = 0; q < 4; ++q) red[q][tid] += red[q][tid + s];
        __syncthreads();
    }
    const float invD = 1.f / (float)D;
    float mean = red[0][0] * invD;
    float mj   = red[1][0] * invD;
    float var  = red[2][0] * invD - mean * mean;
    float mxxj = red[3][0] * invD - mean * mj;
    float corr = 2.f * mxxj;
    float std_ = sqrtf(var + 1e-5f);
    float rstd = 1.f / std_;
    float rstd3 = rstd * rstd * rstd;

    float* outr  = out  + (long)row * D;
    float* outjr = outj + (long)row * D;
#pragma unroll
    for (int t = 0; t < 4; ++t) {
        int i = tid + t * 256;
        float xc   = lx[t] - mean;
        float xhat = xc * rstd;
        float xjc  = lxj[t] - mj;
        float cc   = xjc * rstd - 0.5f * corr * xc * rstd3;
        outr[i]  = xhat * w[i] + b[i];
        outjr[i] = xhat * lw[i] + lb[i] + cc * w[i];
    }
}

// ---------------- softmax over rows (in place) ------------------------------
__global__ __launch_bounds__(256)
void softmax_rows(float* __restrict__ S, int NC)
{
    float* p = S + (long)blockIdx.x * NC;
    const int tid = threadIdx.x;
    __shared__ float red[256];
    float v[4];
    float mx = -3.4e38f;
    for (int i = tid, t = 0; i < NC; i += 256, ++t) { v[t] = p[i]; mx = fmaxf(mx, v[t]); }
    red[tid] = mx; __syncthreads();
    for (int s = 128; s > 0; s >>= 1) { if (tid < s) red[tid] = fmaxf(red[tid], red[tid + s]); __syncthreads(); }
    mx = red[0]; __syncthreads();
    float sum = 0.f;
    for (int i = tid, t = 0; i < NC; i += 256, ++t) { v[t] = expf(v[t] - mx); sum += v[t]; }
    red[tid] = sum; __syncthreads();
    for (int s = 128; s > 0; s >>= 1) { if (tid < s) red[tid] += red[tid + s]; __syncthreads(); }
    float inv = 1.f / red[0];
    for (int i = tid, t = 0; i < NC; i += 256, ++t) p[i] = v[t] * inv;
}

// -------- softmax-JVP filter: filt = (dm - sum(dm*A)) * A  (in place on dm) --
__global__ __launch_bounds__(256)
void attn_filt(float* __restrict__ DM, const float* __restrict__ A, int NC)
{
    float* d = DM + (long)blockIdx.x * NC;
    const float* a = A + (long)blockIdx.x * NC;
    const int tid = threadIdx.x;
    __shared__ float red[256];
    float dv[4], av[4];
    float s = 0.f;
    for (int i = tid, t = 0; i < NC; i += 256, ++t) { dv[t] = d[i]; av[t] = a[i]; s += dv[t] * av[t]; }
    red[tid] = s; __syncthreads();
    for (int st = 128; st > 0; st >>= 1) { if (tid < st) red[tid] += red[tid + st]; __syncthreads(); }
    s = red[0];
    for (int i = tid, t = 0; i < NC; i += 256, ++t) d[i] = (dv[t] - s) * av[t];
}

// ---------------- exact-erf GELU forward + JVP (in place) -------------------
__global__ __launch_bounds__(256)
void gelu_jvp(float* __restrict__ X, float* __restrict__ XJ, long n)
{
    long i = (long)blockIdx.x * blockDim.x + threadIdx.x;
    long stride = (long)gridDim.x * blockDim.x;
    for (; i < n; i += stride) {
        float x = X[i], xj = XJ[i];
        float out = 0.5f * x * (1.f + erff(x * 0.70710678118654752f));
        float pdf = expf(-0.5f * x * x) * 0.39894228040143268f;
        X[i]  = out;
        XJ[i] = xj * (out / (x + 1e-8f) + x * pdf);
    }
}

// ---------------------------------------------------------------------------
static inline void launch_gemm(hipStream_t s, float* C,
                               const float* A1, const float* W1,
                               const float* A2, const float* W2,
                               const float* bias, const float* Res,
                               int M, int Nc, int K,
                               long a_rs, long a_bb, long a_bh,
                               long w_ns, long w_ks, long w_bb, long w_bh,
                               long c_rs, long c_bb, long c_bh,
                               long r_rs, long r_bb, long r_bh,
                               int Z, int Hdiv, float alpha, int a_v4, int w_v4)
{
    dim3 grid((Nc + NT - 1) / NT, (M + MT - 1) / MT, Z);
    if (A2)
        gemm_f32_wmma<true><<<grid, dim3(256), 0, s>>>(C, A1, W1, A2, W2, bias, Res,
            M, Nc, K, a_rs, a_bb, a_bh, w_ns, w_ks, w_bb, w_bh,
            c_rs, c_bb, c_bh, r_rs, r_bb, r_bh, Hdiv, alpha, a_v4, w_v4);
    else
        gemm_f32_wmma<false><<<grid, dim3(256), 0, s>>>(C, A1, W1, A2, W2, bias, Res,
            M, Nc, K, a_rs, a_bb, a_bh, w_ns, w_ks, w_bb, w_bh,
            c_rs, c_bb, c_bh, r_rs, r_bb, r_bh, Hdiv, alpha, a_v4, w_v4);
}

extern "C" void kernel_launch(void* const* d_in, const int* in_sizes, int n_in,
                              void* d_out, int out_size, void* d_ws, size_t ws_size,
                              hipStream_t stream)
{
    (void)in_sizes; (void)n_in; (void)out_size; (void)ws_size;

    const float* input     = (const float*)d_in[0];
    const float* input_jvp = (const float*)d_in[1];
    const float* n1_w  = (const float*)d_in[2];
    const float* n1_b  = (const float*)d_in[3];
    const float* n1_lw = (const float*)d_in[4];
    const float* n1_lb = (const float*)d_in[5];
    const float* qkv_w  = (const float*)d_in[6];
    const float* qkv_lw = (const float*)d_in[7];
    const float* proj_w  = (const float*)d_in[8];
    const float* proj_b  = (const float*)d_in[9];
    const float* proj_lw = (const float*)d_in[10];
    const float* proj_lb = (const float*)d_in[11];
    const float* n2_w  = (const float*)d_in[12];
    const float* n2_b  = (const float*)d_in[13];
    const float* n2_lw = (const float*)d_in[14];
    const float* n2_lb = (const float*)d_in[15];
    const float* fc1_w  = (const float*)d_in[16];
    const float* fc1_b  = (const float*)d_in[17];
    const float* fc1_lw = (const float*)d_in[18];
    const float* fc1_lb = (const float*)d_in[19];
    const float* fc2_w  = (const float*)d_in[20];
    const float* fc2_b  = (const float*)d_in[21];
    const float* fc2_lw = (const float*)d_in[22];
    const float* fc2_lb = (const float*)d_in[23];

    const int  M  = MROWS;                      // 3140
    const long Md = (long)M * DMODEL;
    const long NN = (long)NSEQ * NSEQ;          // 616225
    const long ZATT = (long)BDIM * NH;          // 64

    float* out_x  = (float*)d_out;
    float* out_xj = out_x + Md;

    // workspace carve-up (fp32)
    float* ws    = (float*)d_ws;
    float* x1    = ws;                      // [M,1024]
    float* x1j   = x1   + Md;
    float* qkv   = x1j  + Md;               // [M,3072]
    float* qkvj  = qkv  + (long)M * 3072;
    float* attnA = qkvj + (long)M * 3072;   // [64,785,785]
    float* dmF   = attnA + ZATT * NN;       // [64,785,785]
    float* aout  = dmF  + ZATT * NN;        // [M,1024]
    float* aoutj = aout + Md;
    float* x2    = aoutj + Md;              // residual-1 result
    float* x2j   = x2   + Md;
    float* z2    = x2j  + Md;               // LN2 out
    float* z2j   = z2   + Md;
    float* hbuf  = z2j  + Md;               // [M,4096]
    float* hjbuf = hbuf + (long)M * HID;

    const float scale = 0.125f;             // 64^-0.5

    // ---- 1. LN1 ----
    ln_jvp<<<dim3(M), dim3(256), 0, stream>>>(input, input_jvp, n1_w, n1_b, n1_lw, n1_lb, x1, x1j);

    // ---- 2. qkv = x1 @ qkv_w^T ; qkvj = x1j @ qkv_w^T + x1 @ qkv_lw^T (no bias) ----
    launch_gemm(stream, qkv,  x1,  qkv_w, nullptr, nullptr, nullptr, nullptr,
                M, 3 * DMODEL, DMODEL,
                DMODEL, 0, 0,  DMODEL, 1, 0, 0,  3 * DMODEL, 0, 0,  0, 0, 0, 1, 1, 1.f, 1, 1);
    launch_gemm(stream, qkvj, x1j, qkv_w, x1, qkv_lw, nullptr, nullptr,
                M, 3 * DMODEL, DMODEL,
                DMODEL, 0, 0,  DMODEL, 1, 0, 0,  3 * DMODEL, 0, 0,  0, 0, 0, 1, 1, 1.f, 1, 1);

    // ---- 3. scores S = scale * q k^T  (batched over z = b*16 + h) ----
    const long qkv_bb = (long)NSEQ * 3072;  // batch stride in qkv
    launch_gemm(stream, attnA,
                qkv /*q*/, qkv + 1024 /*k*/, nullptr, nullptr, nullptr, nullptr,
                NSEQ, NSEQ, HD,
                3072, qkv_bb, 64,
                3072, 1, qkv_bb, 64,
                NSEQ, (long)NH * NN, NN,
                0, 0, 0, (int)ZATT, NH, scale, 1, 1);
    // ---- 4. dm = scale * (qj k^T + q kj^T) ----
    launch_gemm(stream, dmF,
                qkvj /*qj*/, qkv + 1024 /*k*/, qkv /*q*/, qkvj + 1024 /*kj*/,
                nullptr, nullptr,
                NSEQ, NSEQ, HD,
                3072, qkv_bb, 64,
                3072, 1, qkv_bb, 64,
                NSEQ, (long)NH * NN, NN,
                0, 0, 0, (int)ZATT, NH, scale, 1, 1);

    // ---- 5. softmax rows of S -> A ; filt = (dm - sum(dm*A))*A ----
    softmax_rows<<<dim3((unsigned)(ZATT * NSEQ)), dim3(256), 0, stream>>>(attnA, NSEQ);
    attn_filt  <<<dim3((unsigned)(ZATT * NSEQ)), dim3(256), 0, stream>>>(dmF, attnA, NSEQ);

    // ---- 6. out = A @ v ; out_j = filt @ v + A @ vj   (NN gemm: w_ks = 3072) ----
    launch_gemm(stream, aout,
                attnA, qkv + 2048 /*v*/, nullptr, nullptr, nullptr, nullptr,
                NSEQ, HD, NSEQ,
                NSEQ, (long)NH * NN, NN,
                1, 3072, qkv_bb, 64,
                DMODEL, (long)NSEQ * DMODEL, 64,
                0, 0, 0, (int)ZATT, NH, 1.f, 0, 0);
    launch_gemm(stream, aoutj,
                dmF /*filt*/, qkv + 2048 /*v*/, attnA, qkvj + 2048 /*vj*/,
                nullptr, nullptr,
                NSEQ, HD, NSEQ,
                NSEQ, (long)NH * NN, NN,
                1, 3072, qkv_bb, 64,
                DMODEL, (long)NSEQ * DMODEL, 64,
                0, 0, 0, (int)ZATT, NH, 1.f, 0, 0);

    // ---- 7. proj (+bias) fused with residual: x2 = aout@proj_w^T + proj_b + input ----
    launch_gemm(stream, x2,  aout,  proj_w, nullptr, nullptr, proj_b, input,
                M, DMODEL, DMODEL,
                DMODEL, 0, 0,  DMODEL, 1, 0, 0,  DMODEL, 0, 0,  DMODEL, 0, 0, 1, 1, 1.f, 1, 1);
    launch_gemm(stream, x2j, aoutj, proj_w, aout, proj_lw, proj_lb, input_jvp,
                M, DMODEL, DMODEL,
                DMODEL, 0, 0,  DMODEL, 1, 0, 0,  DMODEL, 0, 0,  DMODEL, 0, 0, 1, 1, 1.f, 1, 1);

    // ---- 8. LN2 ----
    ln_jvp<<<dim3(M), dim3(256), 0, stream>>>(x2, x2j, n2_w, n2_b, n2_lw, n2_lb, z2, z2j);

    // ---- 9. fc1 ----
    launch_gemm(stream, hbuf,  z2,  fc1_w, nullptr, nullptr, fc1_b, nullptr,
                M, HID, DMODEL,
                DMODEL, 0, 0,  DMODEL, 1, 0, 0,  HID, 0, 0,  0, 0, 0, 1, 1, 1.f, 1, 1);
    launch_gemm(stream, hjbuf, z2j, fc1_w, z2, fc1_lw, fc1_lb, nullptr,
                M, HID, DMODEL,
                DMODEL, 0, 0,  DMODEL, 1, 0, 0,  HID, 0, 0,  0, 0, 0, 1, 1, 1.f, 1, 1);

    // ---- 10. GELU + JVP in place ----
    gelu_jvp<<<dim3(8192), dim3(256), 0, stream>>>(hbuf, hjbuf, (long)M * HID);

    // ---- 11. fc2 fused with residual-2, straight into d_out ----
    launch_gemm(stream, out_x,  hbuf,  fc2_w, nullptr, nullptr, fc2_b, x2,
                M, DMODEL, HID,
                HID, 0, 0,  HID, 1, 0, 0,  DMODEL, 0, 0,  DMODEL, 0, 0, 1, 1, 1.f, 1, 1);
    launch_gemm(stream, out_xj, hjbuf, fc2_w, hbuf, fc2_lw, fc2_lb, x2j,
                M, DMODEL, HID,
                HID, 0, 0,  HID, 1, 0, 0,  DMODEL, 0, 0,  DMODEL, 0, 0, 1, 1, 1.f, 1, 1);
}